// Laplacian_Forward_Model_27616639714072
// MI455X (gfx1250) — compile-verified
//
#include <hip/hip_runtime.h>

typedef __attribute__((ext_vector_type(2))) float v2f;
typedef __attribute__((ext_vector_type(8))) float v8f;

// Precomputed B-matrix WMMA fragments (image-invariant, built once per launch):
// layout: [J(2)][qpair i(12)][lane(32)] -> float4 = (q=2i,e0),(q=2i,e1),(q=2i+1,e0),(q=2i+1,e1)
__device__ float4 g_btab[768];

// Btil[32*blk + ii][n] = ker[blk][ii - n + 1] if |ii-n|<=1 else 0
__global__ void build_btab_kernel(const float* __restrict__ g_w,
                                  const float* __restrict__ g_wf) {
    int f = blockIdx.x * 256 + threadIdx.x;   // 0..3071
    if (f >= 3072) return;

    float wf = g_wf[0];
    wf = fminf(fmaxf(wf, 1.001f), 254.999f);
    float k0 = fminf(fmaxf(g_w[0], -0.999f), 0.999f) * wf;
    float k1 = fminf(fmaxf(g_w[1], -0.999f), 0.999f) * wf;
    float k2 = fminf(fmaxf(g_w[2], -0.999f), 0.999f) * wf;
    float k3 = fminf(fmaxf(g_w[3], -0.999f), 0.999f) * wf;
    float k4 = fminf(fmaxf(g_w[4], -0.999f), 0.999f) * wf;
    float k5 = fminf(fmaxf(g_w[5], -0.999f), 0.999f) * wf;
    float k6 = fminf(fmaxf(g_w[6], -0.999f), 0.999f) * wf;
    float k7 = fminf(fmaxf(g_w[7], -0.999f), 0.999f) * wf;
    float k8 = fminf(fmaxf(g_w[8], -0.999f), 0.999f) * wf;

    const int j    = f & 3;          // component within float4
    const int lane = (f >> 2) & 31;
    const int rest = f >> 7;         // 0..23
    const int i    = rest % 12;      // q-pair
    const int J    = rest / 12;      // output col tile
    const int q    = 2 * i + (j >> 1);
    const int e    = j & 1;
    const int hi   = lane >> 4;
    const int n    = 16 * J + (lane & 15);
    const int c    = 4 * q + 2 * hi + e;
    const int blk  = c >> 5;
    const int d    = (c & 31) - n;

    const float ka = (blk == 0) ? k0 : (blk == 1) ? k3 : k6;
    const float kb = (blk == 0) ? k1 : (blk == 1) ? k4 : k7;
    const float kc = (blk == 0) ? k2 : (blk == 1) ? k5 : k8;
    float val = 0.0f;
    val = (d == -1) ? ka : val;
    val = (d ==  0) ? kb : val;
    val = (d ==  1) ? kc : val;
    ((float*)g_btab)[f] = val;
}

// One workgroup (128 threads = 4 waves) per 32x32 image.
// Wave w computes out tile (I=w>>1, J=w&1): Out = Atil(32x96)*Btil(96x32)
// via 24 x V_WMMA_F32_16X16X4_F32 — exact f32 3x3 zero-padded convolution.
__global__ __launch_bounds__(128) void lap_sharp_wmma_kernel(
    const float* __restrict__ g_in,
    float* __restrict__ g_out)
{
    // padded image rows -1..32 stored as rows 0..33, stride 36 floats
    // (stride 36: 16B-aligned float4 rows, 16 distinct banks over 16 rows)
    __shared__ float s_img[34 * 36];
    __shared__ float s_wmax[4], s_wmin[4];
    __shared__ float s_wmax2[4], s_wmin2[4];

    const int t        = threadIdx.x;
    const int img      = blockIdx.x;
    const int lane     = t & 31;
    const int wave     = t >> 5;
    const int laneHalf = lane & 15;
    const int hi       = (lane >> 4) & 1;
    const int I        = wave >> 1;
    const int J        = wave & 1;

    // zero halo rows (-1 and 32)
    if (t < 36) {
        s_img[t]           = 0.0f;
        s_img[33 * 36 + t] = 0.0f;
    }
    // image -> LDS: 2 x float4 per thread, aligned b128 LDS stores
    const float4* src4 = (const float4*)(g_in + (size_t)img * 1024);
#pragma unroll
    for (int k = 0; k < 2; ++k) {
        const int i4 = k * 128 + t;          // float4 index 0..255
        const int y  = i4 >> 3;
        const int x4 = (i4 & 7) << 2;
        *(float4*)&s_img[(y + 1) * 36 + x4] = src4[i4];
    }

    // B fragments from the precomputed table (hits WGP$/L2)
    float4 bt[12];
#pragma unroll
    for (int i = 0; i < 12; ++i)
        bt[i] = g_btab[(J * 12 + i) * 32 + lane];

    __syncthreads();

    // ---- conv via WMMA ----
    const int yA = 16 * I + laneHalf;
    v8f acc = {};
#pragma unroll
    for (int q = 0; q < 24; ++q) {
        const int c0 = 4 * q + 2 * hi;
        const int c1 = c0 + 1;
        v2f a;
        a.x = s_img[(yA + (c0 >> 5)) * 36 + (c0 & 31)];
        a.y = s_img[(yA + (c1 >> 5)) * 36 + (c1 & 31)];
        const float4 fb = bt[q >> 1];
        v2f b;
        b.x = (q & 1) ? fb.z : fb.x;
        b.y = (q & 1) ? fb.w : fb.y;
        acc = __builtin_amdgcn_wmma_f32_16x16x4_f32(
            false, a, false, b, (short)0, acc, false, false);
    }

    // this lane's conv values: row = 16I + v + 8*hi, col = 16J + laneHalf
    float conv[8];
    float lmax = -3.402823466e38f, lmin = 3.402823466e38f;
#pragma unroll
    for (int v = 0; v < 8; ++v) {
        conv[v] = acc[v];
        lmax = fmaxf(lmax, conv[v]);
        lmin = fminf(lmin, conv[v]);
    }

    // ---- reduction 1: wave32 butterfly + 4-entry LDS combine ----
#pragma unroll
    for (int m = 16; m >= 1; m >>= 1) {
        lmax = fmaxf(lmax, __shfl_xor(lmax, m, 32));
        lmin = fminf(lmin, __shfl_xor(lmin, m, 32));
    }
    if (lane == 0) { s_wmax[wave] = lmax; s_wmin[wave] = lmin; }
    __syncthreads();
    const float cmax = fmaxf(fmaxf(s_wmax[0], s_wmax[1]), fmaxf(s_wmax[2], s_wmax[3]));
    const float cmin = fminf(fminf(s_wmin[0], s_wmin[1]), fminf(s_wmin[2], s_wmin[3]));

    const float denom1 = fmaxf(cmax - cmin, 1e-6f);
    const int col = 16 * J + laneHalf;
    float sharp[8];
    lmax = -3.402823466e38f; lmin = 3.402823466e38f;
#pragma unroll
    for (int v = 0; v < 8; ++v) {
        const int y = 16 * I + v + 8 * hi;
        const float xv = s_img[(y + 1) * 36 + col];
        const float sh = xv + (conv[v] - cmin) / denom1 * 255.0f;
        sharp[v] = sh;
        lmax = fmaxf(lmax, sh);
        lmin = fminf(lmin, sh);
    }

    // ---- reduction 2 ----
#pragma unroll
    for (int m = 16; m >= 1; m >>= 1) {
        lmax = fmaxf(lmax, __shfl_xor(lmax, m, 32));
        lmin = fminf(lmin, __shfl_xor(lmin, m, 32));
    }
    if (lane == 0) { s_wmax2[wave] = lmax; s_wmin2[wave] = lmin; }
    __syncthreads();
    const float smax = fmaxf(fmaxf(s_wmax2[0], s_wmax2[1]), fmaxf(s_wmax2[2], s_wmax2[3]));
    const float smin = fminf(fminf(s_wmin2[0], s_wmin2[1]), fminf(s_wmin2[2], s_wmin2[3]));

    const float denom2 = fminf(fmaxf(smax - smin, 1e-6f), 255.0f);
    float* dst = g_out + (size_t)img * 1024;
#pragma unroll
    for (int v = 0; v < 8; ++v) {
        const int y = 16 * I + v + 8 * hi;
        float r = (sharp[v] - smin) / denom2 * 255.0f;
        r = fminf(fmaxf(r, 0.0f), 255.0f);
        dst[y * 32 + col] = floorf(r);
    }
}

extern "C" void kernel_launch(void* const* d_in, const int* in_sizes, int n_in,
                              void* d_out, int out_size, void* d_ws, size_t ws_size,
                              hipStream_t stream) {
    const float* g_in = (const float*)d_in[0];   // (B,1,32,32) f32
    const float* g_w  = (const float*)d_in[1];   // (1,9)       f32
    const float* g_wf = (const float*)d_in[2];   // (1,1)       f32
    float* g_out      = (float*)d_out;           // (1,B,1,32,32) f32

    build_btab_kernel<<<12, 256, 0, stream>>>(g_w, g_wf);
    const int nimg = in_sizes[0] >> 10;          // B = elems / 1024
    lap_sharp_wmma_kernel<<<nimg, 128, 0, stream>>>(g_in, g_out);
}